// VerifyConv2d_38482906972593
// MI455X (gfx1250) — compile-verified
//
#include <hip/hip_runtime.h>

// VerifyConv2d backsubstitution on gfx1250 (MI455X), f32 WMMA path, round 2.
//
// new_uc = uc @ T (conv Toeplitz) == transposed 3x3 conv; same for lc.
// Tile: D[ci(16) x 16 iw] += A[ci x K] * B[K x iw], K = 288 = 72 x 4,
// k = (kh*3+kw)*32 + co, via V_WMMA_F32_16X16X4_F32 (exact f32; problem is
// HBM-bound at ~25MB so the slow f32 WMMA rate is irrelevant).
//
// Round-2 fixes vs round-1:
//  * wave id forced scalar (readfirstlane) -> ih loop is uniform, EXEC==~0
//    around every WMMA (ISA 7.12 requirement).
//  * B operand gathered from a zero-padded LDS slab [ih+2][iw+2][co] ->
//    no border masks, no saveexec-predicated loads; inner loop is
//    ds_load_b64(A) + ds_load_b64(B) + v_wmma, all immediate-offset.
//  * co dimension padded to 34 dwords -> conflict-free ds_load_b64
//    (each 16-lane half covers 32 distinct banks of 64).
//  * dual accumulators to decouple the WMMA RAW chain.

typedef __attribute__((ext_vector_type(2))) float v2f;
typedef __attribute__((ext_vector_type(4))) float v4f;
typedef __attribute__((ext_vector_type(8))) float v8f;

#define CIN      16
#define COUT     32
#define HH       16
#define WW       16
#define OHW      256      // H*W per channel plane
#define IN_DIM   4096     // CIN*H*W
#define OUT_DIM  8192     // COUT*OH*OW
#define NROWS    256      // constraint rows
#define NSTEPS   72       // 288 / K(=4)

#define PH       18       // padded height
#define PW       18       // padded width
#define PC       34       // padded channel stride (dwords), bank-conflict-free
#define UB_DW    (PH * PW * PC)          // 11016 dwords used
#define UB_ALLOC 11264                   // 44 * 128 * 2 dwords, uniform zero loop

__global__ __launch_bounds__(128) void conv_backsub_wmma(
    const float* __restrict__ uc, const float* __restrict__ lc,
    const float* __restrict__ weight, float* __restrict__ out)
{
  // A fragments in native wave32 A-matrix layout:
  //   lane l -> M = ci = l&15 ; VGPR v -> K = kbase + 2*(l>>4) + v
  __shared__ v2f  Afrag[NSTEPS][32];
  // Zero-padded activation slab, layout [ih+1][iw+1][co], co-stride PC.
  __shared__ float uB[UB_ALLOC];

  const int tid  = threadIdx.x;
  const int lane = tid & 31;
  const int wave = __builtin_amdgcn_readfirstlane(tid >> 5);  // scalar!
  const int l16  = lane & 15;   // A: ci row | B/D: iw column
  const int half = lane >> 4;

  // ---- zero the slab (uniform trip count: 44 iters * 128 thr * v2f) ------
  v2f z2; z2.x = 0.f; z2.y = 0.f;
#pragma unroll 4
  for (int i = tid; i < UB_ALLOC / 2; i += 128)
    ((v2f*)uB)[i] = z2;

  // ---- stage Toeplitz weight A[ci][k], k = (kh*3+kw)*32 + co -------------
  for (int s = wave; s < NSTEPS; s += 4) {          // uniform per wave
    const int g  = s >> 3;                          // kh*3+kw
    const int co = ((s & 7) << 2) + (half << 1);
    v2f a;
    a.x = weight[(co + 0) * (CIN * 9) + l16 * 9 + g];
    a.y = weight[(co + 1) * (CIN * 9) + l16 * 9 + g];
    Afrag[s][lane] = a;
  }

  const int bid = blockIdx.x;                       // 0..511
  const int m   = bid >> 8;                         // 0 = uc, 1 = lc
  const int n   = bid & 255;
  const float* __restrict__ src = (m ? lc : uc) + n * OUT_DIM;
  float* __restrict__ dst = out + m * (NROWS * IN_DIM) + n * IN_DIM;

  __syncthreads();

  // ---- fill interior: src[co][sp] -> uB[ih+1][iw+1][co] ------------------
  // 16 uniform iterations; global reads are b128-coalesced.
#pragma unroll 4
  for (int i = tid; i < OUT_DIM / 4; i += 128) {
    const int base = i << 2;                        // element index, iw%4==0
    const int co   = base >> 8;
    const int sp   = base & 255;
    const int ihh  = sp >> 4;
    const int iww  = sp & 15;
    const v4f v = *(const v4f*)(src + base);
    float* p = &uB[((ihh + 1) * PW + (iww + 1)) * PC + co];
    p[0 * PC] = v.x; p[1 * PC] = v.y; p[2 * PC] = v.z; p[3 * PC] = v.w;
  }
  __syncthreads();

  // ---- main GEMM: each wave owns 4 output rows (ih) ----------------------
  for (int ih = wave; ih < HH; ih += 4) {           // uniform loop
    v8f acc0 = {0.f, 0.f, 0.f, 0.f, 0.f, 0.f, 0.f, 0.f};
    v8f acc1 = {0.f, 0.f, 0.f, 0.f, 0.f, 0.f, 0.f, 0.f};
    const int co2 = half << 1;                      // 2*half

#pragma unroll
    for (int kh = 0; kh < 3; ++kh) {
      const int ihr = ih + 2 - kh;                  // padded row, always valid
#pragma unroll
      for (int kw = 0; kw < 3; ++kw) {
        const int iwr = l16 + 2 - kw;               // padded col, always valid
        const float* __restrict__ bbase =
            &uB[(ihr * PW + iwr) * PC + co2];       // + cb*4 (imm offset)
        const int g = kh * 3 + kw;
#pragma unroll
        for (int cb = 0; cb < 8; ++cb) {            // 32 co in chunks of K=4
          const v2f a = Afrag[g * 8 + cb][lane];    // ds_load_b64, imm offset
          const v2f b = *(const v2f*)(bbase + (cb << 2)); // ds_load_b64
          if (cb & 1)
            acc1 = __builtin_amdgcn_wmma_f32_16x16x4_f32(
                false, a, false, b, (short)0, acc1, false, false);
          else
            acc0 = __builtin_amdgcn_wmma_f32_16x16x4_f32(
                false, a, false, b, (short)0, acc0, false, false);
        }
      }
    }

    // D layout: lane l, VGPR v -> (ci = v + 8*half, iw = l16)
    const int sp = ih * WW + l16;
#pragma unroll
    for (int v = 0; v < 8; ++v)
      dst[(half * 8 + v) * OHW + sp] = acc0[v] + acc1[v];
  }
}

// ---- bias backsubstitution: one wave per (matrix, n) ----------------------
__global__ __launch_bounds__(32) void bias_backsub(
    const float* __restrict__ uc, const float* __restrict__ lc,
    const float* __restrict__ ucb, const float* __restrict__ lcb,
    const float* __restrict__ bias, float* __restrict__ out)
{
  const int bid = blockIdx.x;                       // 0..511
  const int m   = bid >> 8;
  const int n   = bid & 255;
  const float* __restrict__ src = (m ? lc : uc) + n * OUT_DIM;
  const float* __restrict__ bin = m ? lcb : ucb;

  const int lane = threadIdx.x;
  float p = 0.f;
  for (int j = lane; j < OUT_DIM; j += 32)
    p += src[j] * bias[j >> 8];                     // bvec = repeat(bias,256)

#pragma unroll
  for (int off = 16; off > 0; off >>= 1)
    p += __shfl_xor(p, off, 32);

  if (lane == 0)
    out[2 * NROWS * IN_DIM + m * NROWS + n] = bin[n] + p;
}

extern "C" void kernel_launch(void* const* d_in, const int* in_sizes, int n_in,
                              void* d_out, int out_size, void* d_ws, size_t ws_size,
                              hipStream_t stream) {
  (void)in_sizes; (void)n_in; (void)out_size; (void)d_ws; (void)ws_size;
  const float* uc     = (const float*)d_in[0];
  const float* lc     = (const float*)d_in[1];
  const float* ucb    = (const float*)d_in[2];
  const float* lcb    = (const float*)d_in[3];
  const float* weight = (const float*)d_in[4];
  const float* bias   = (const float*)d_in[5];
  float* out = (float*)d_out;

  conv_backsub_wmma<<<512, 128, 0, stream>>>(uc, lc, weight, out);
  bias_backsub<<<512, 32, 0, stream>>>(uc, lc, ucb, lcb, bias, out);
}